// MLPProjectionFilter_18305150615979
// MI455X (gfx1250) — compile-verified
//
#include <hip/hip_runtime.h>
#include <math.h>

// Problem constants (from reference)
#define BATCH   2048
#define NVAR    192          // DOF * NVS
#define NCON    1200         // 2 * NUM * DOF
#define NEQ     30
#define NQ      222          // NVAR + NEQ
#define KPAD    224          // NQ padded to multiple of 4
#define MAXITER 15
#define TM      16           // batch rows per workgroup (= WMMA M)
#define NWAVES  12
#define NTHREADS (NWAVES*32)
#define V_MAXC  0.8f
#define A_MAXC  1.8f
#define P_MAXC  3.14159265358979323846f
#define NTILE_C (NCON/TM)    // 75

// ---- LDS layout: row-major per batch-row, padded strides (bank-friendly,
//      8B aligned so WMMA K-pair fragments are single ds_load_b64) ----
#define AVX_RS  3620         // sAvx row stride (3*1204 + pad), 3620%64=36
#define AVX_TS  1204         // per-constraint-set offset inside a row
#define C_RS    196          // 196%64=4
#define RHS_RS  228          // 228%64=36
#define L_RS    192          // elementwise only

#define OFF_AVX 0
#define OFF_C   (OFF_AVX + TM*AVX_RS)   // 57920
#define OFF_RHS (OFF_C   + TM*C_RS)     // 61056
#define OFF_L   (OFF_RHS + TM*RHS_RS)   // 64704
#define OFF_N   (OFF_L   + TM*L_RS)     // 67776  [10][16] norm sums
#define OFF_FP  (OFF_N   + 10*TM)
#define OFF_RP  (OFF_FP  + TM)
#define LDS_FLOATS (OFF_RP + TM)        // 67968 floats = 271,872 B (< 320 KB)

typedef __attribute__((ext_vector_type(2))) float v2f;
typedef __attribute__((ext_vector_type(8))) float v8f;

__device__ __forceinline__ v8f wmma_f32(v2f a, v2f b, v8f c) {
  // D(16x16,f32) = A(16x4,f32) * B(4x16,f32) + C
  return __builtin_amdgcn_wmma_f32_16x16x4_f32(false, a, false, b, (short)0, c,
                                               false, false);
}

// ---- one constraint set of the fused lincost/lambda pass ----
// accg += min(b, Avx) @ A   (g, for -lincost)
// hacc += relu(Avx - b) @ A (h, lambda update + fixed-point norm)
template<bool WANT_G, bool WANT_H>
__device__ __forceinline__ void lincost_pass(const float* __restrict__ A,
                                             float bval, int tkofs,
                                             int n0, int lm, int lhi,
                                             v8f& accg, v8f& hacc) {
  extern __shared__ float lds[];
  unsigned ia = (unsigned)(2*lhi)*NVAR + (unsigned)(n0 + lm);      // global A
  unsigned ix = OFF_AVX + (unsigned)lm*AVX_RS + (unsigned)tkofs + 2*lhi;
  #pragma unroll 4
  for (int k0 = 0; k0 < NCON; k0 += 4) {
    v2f bf;
    bf[0] = A[ia];             // rows 64B-contiguous across lanes (saddr form)
    bf[1] = A[ia + NVAR];
    v2f x = *(const v2f*)&lds[ix];   // one ds_load_b64 per lane
    if (WANT_G) {
      v2f mf; mf[0] = fminf(bval, x[0]); mf[1] = fminf(bval, x[1]);
      accg = wmma_f32(mf, bf, accg);
    }
    if (WANT_H) {
      v2f rf; rf[0] = fmaxf(x[0] - bval, 0.0f); rf[1] = fmaxf(x[1] - bval, 0.0f);
      hacc = wmma_f32(rf, bf, hacc);
    }
    ia += 4*NVAR;
    ix += 4;
  }
}

// ---- step A: rhs = L_new + g (= -lincost); L -= sum h; ||h_k||^2 norms ----
template<bool WANT_G, bool WANT_H>
__device__ __forceinline__ void step_lincost(const float* __restrict__ Av,
                                             const float* __restrict__ Aa,
                                             const float* __restrict__ Ap,
                                             int wave, int lm, int lhi) {
  extern __shared__ float lds[];
  const int n0 = wave * TM;
  v8f accg = {};
  v8f h0 = {}, h1 = {}, h2 = {};
  lincost_pass<WANT_G, WANT_H>(Av, V_MAXC, 0*AVX_TS, n0, lm, lhi, accg, h0);
  lincost_pass<WANT_G, WANT_H>(Aa, A_MAXC, 1*AVX_TS, n0, lm, lhi, accg, h1);
  lincost_pass<WANT_G, WANT_H>(Ap, P_MAXC, 2*AVX_TS, n0, lm, lhi, accg, h2);
  #pragma unroll
  for (int e = 0; e < 8; ++e) {
    int m = e + 8*lhi;
    int n = n0 + lm;
    float hv = h0[e], ha = h1[e], hp = h2[e];
    float Ln = lds[OFF_L + m*L_RS + n] - hv - ha - hp;  // h==0 when !WANT_H
    lds[OFF_L + m*L_RS + n] = Ln;
    if (WANT_G) lds[OFF_RHS + m*RHS_RS + n] = Ln + accg[e];
    if (WANT_H) {
      atomicAdd(&lds[OFF_N + 7*TM + m], hv*hv);
      atomicAdd(&lds[OFF_N + 8*TM + m], ha*ha);
      atomicAdd(&lds[OFF_N + 9*TM + m], hp*hp);
    }
  }
}

// ---- step B: primal = rhs @ Qinv_top^T; dc norm; c_prev <- primal ----
__device__ __forceinline__ void step_primal(const float* __restrict__ Qinv,
                                            int wave, int lm, int lhi) {
  extern __shared__ float lds[];
  const int n0 = wave * TM;
  v8f acc = {};
  unsigned iq = (unsigned)(n0 + lm)*NQ + 2*lhi;
  unsigned ir = OFF_RHS + (unsigned)lm*RHS_RS + 2*lhi;
  #pragma unroll 4
  for (int k0 = 0; k0 < KPAD; k0 += 4) {
    int k = k0 + 2*lhi;
    v2f bf = *(const v2f*)&Qinv[iq];         // global_load_b64, 8B aligned
    if (k + 0 >= NQ) bf[0] = 0.0f;           // zero the K padding
    if (k + 1 >= NQ) bf[1] = 0.0f;
    v2f af = *(const v2f*)&lds[ir];          // ds_load_b64
    acc = wmma_f32(af, bf, acc);
    iq += 4;
    ir += 4;
  }
  #pragma unroll
  for (int e = 0; e < 8; ++e) {
    int m = e + 8*lhi, n = n0 + lm;
    unsigned ic = OFF_C + (unsigned)m*C_RS + n;
    float p = acc[e];
    float d = p - lds[ic];
    atomicAdd(&lds[OFF_N + 6*TM + m], d*d);
    lds[ic] = p;
  }
}

// ---- step D (one constraint set): Avx = primal @ A^T; res / s-diff norms ----
template<bool INIT>
__device__ __forceinline__ void step_avx(const float* __restrict__ A, float bval,
                                         int tk, int wave, int lm, int lhi) {
  extern __shared__ float lds[];
  for (int ct = wave; ct < NTILE_C; ct += NWAVES) {
    const int c0 = ct * TM;
    v8f acc = {};
    unsigned ia = (unsigned)(c0 + lm)*NVAR + 2*lhi;       // A^T[k][n] = A[n][k]
    unsigned ic = OFF_C + (unsigned)lm*C_RS + 2*lhi;
    #pragma unroll 8
    for (int k0 = 0; k0 < NVAR; k0 += 4) {
      v2f af = *(const v2f*)&lds[ic];        // ds_load_b64 (primal K-pair)
      v2f bf = *(const v2f*)&A[ia];          // global_load_b64 (A row K-pair)
      acc = wmma_f32(af, bf, acc);
      ia += 4;
      ic += 4;
    }
    #pragma unroll
    for (int e = 0; e < 8; ++e) {
      int m = e + 8*lhi;
      unsigned ax = OFF_AVX + (unsigned)m*AVX_RS + (unsigned)(tk*AVX_TS + c0 + lm);
      float x = acc[e];
      if (!INIT) {
        float xo  = lds[ax];
        float res = fmaxf(x - bval, 0.0f);
        float sd  = fmaxf(bval - x, 0.0f) - fmaxf(bval - xo, 0.0f);
        atomicAdd(&lds[OFF_N + tk*TM + m], res*res);
        atomicAdd(&lds[OFF_N + (3 + tk)*TM + m], sd*sd);
      }
      lds[ax] = x;
    }
  }
}

__global__ __launch_bounds__(NTHREADS) void
admm_proj_kernel(const float* __restrict__ lamda, const float* __restrict__ cin,
                 const float* __restrict__ csamp, const float* __restrict__ beq,
                 const float* __restrict__ Av, const float* __restrict__ Aa,
                 const float* __restrict__ Ap, const float* __restrict__ Qinv,
                 float* __restrict__ out) {
  extern __shared__ float lds[];
  const int tid  = threadIdx.x;
  const int wave = tid >> 5;
  const int lane = tid & 31;
  const int lm   = lane & 15;    // A-op M row / B-op N col (per ISA layouts)
  const int lhi  = lane >> 4;    // selects K pair {2*lhi, 2*lhi+1}
  const int rowBase = blockIdx.x * TM;

  // ---------------- init ----------------
  for (int idx = tid; idx < NVAR*TM; idx += NTHREADS) {
    int n = idx >> 4, m = idx & 15, r = rowBase + m;
    lds[OFF_L + m*L_RS + n] =
        lamda[(size_t)r*3*NVAR + n] + lamda[(size_t)r*3*NVAR + NVAR + n] +
        lamda[(size_t)r*3*NVAR + 2*NVAR + n] + cin[(size_t)r*NVAR + n];
    lds[OFF_C + m*C_RS + n] = csamp[(size_t)r*NVAR + n];
  }
  for (int idx = tid; idx < (KPAD - NVAR)*TM; idx += NTHREADS) {
    int k = idx >> 4, m = idx & 15;
    lds[OFF_RHS + m*RHS_RS + NVAR + k] =
        (k < NEQ) ? beq[(size_t)(rowBase + m)*NEQ + k] : 0.0f;
  }
  for (int idx = tid; idx < 12*TM; idx += NTHREADS) lds[OFF_N + idx] = 0.0f;
  __syncthreads();

  // Avx_init = c_samples @ A_k^T
  step_avx<true>(Av, V_MAXC, 0, wave, lm, lhi);
  step_avx<true>(Aa, A_MAXC, 1, wave, lm, lhi);
  step_avx<true>(Ap, P_MAXC, 2, wave, lm, lhi);
  __syncthreads();

  for (int iter = 0; iter < MAXITER; ++iter) {
    if (iter == 0) step_lincost<true, false>(Av, Aa, Ap, wave, lm, lhi);
    else           step_lincost<true, true >(Av, Aa, Ap, wave, lm, lhi);
    __syncthreads();
    if (iter > 0 && tid < TM) {   // fold lambda-diff norms of iteration iter-1
      lds[OFF_FP + tid] += sqrtf(lds[OFF_N + 7*TM + tid]) +
                           sqrtf(lds[OFF_N + 8*TM + tid]) +
                           sqrtf(lds[OFF_N + 9*TM + tid]);
      lds[OFF_N + 7*TM + tid] = 0.0f;
      lds[OFF_N + 8*TM + tid] = 0.0f;
      lds[OFF_N + 9*TM + tid] = 0.0f;
    }
    __syncthreads();
    step_primal(Qinv, wave, lm, lhi);
    __syncthreads();
    step_avx<false>(Av, V_MAXC, 0, wave, lm, lhi);
    step_avx<false>(Aa, A_MAXC, 1, wave, lm, lhi);
    step_avx<false>(Ap, P_MAXC, 2, wave, lm, lhi);
    __syncthreads();
    if (tid < TM) {               // fold res / s-diff / dc norms of this iter
      lds[OFF_RP + tid] += sqrtf(lds[OFF_N + 0*TM + tid]) +
                           sqrtf(lds[OFF_N + 1*TM + tid]) +
                           sqrtf(lds[OFF_N + 2*TM + tid]);
      lds[OFF_FP + tid] += sqrtf(lds[OFF_N + 3*TM + tid]) +
                           sqrtf(lds[OFF_N + 4*TM + tid]) +
                           sqrtf(lds[OFF_N + 5*TM + tid]) +
                           sqrtf(lds[OFF_N + 6*TM + tid]);
      #pragma unroll
      for (int q = 0; q < 7; ++q) lds[OFF_N + q*TM + tid] = 0.0f;
    }
    __syncthreads();
  }

  // h-only pass for the last iteration's lambda-diff norms
  step_lincost<false, true>(Av, Aa, Ap, wave, lm, lhi);
  __syncthreads();
  if (tid < TM)
    lds[OFF_FP + tid] += sqrtf(lds[OFF_N + 7*TM + tid]) +
                         sqrtf(lds[OFF_N + 8*TM + tid]) +
                         sqrtf(lds[OFF_N + 9*TM + tid]);
  __syncthreads();

  // ---------------- outputs: [c_out | avg_fp | avg_rp] ----------------
  for (int idx = tid; idx < NVAR*TM; idx += NTHREADS) {
    int n = idx >> 4, m = idx & 15;
    out[(size_t)(rowBase + m)*NVAR + n] = lds[OFF_C + m*C_RS + n];
  }
  if (tid < TM) {
    out[(size_t)BATCH*NVAR + rowBase + tid]         = lds[OFF_FP + tid] * (1.0f/MAXITER);
    out[(size_t)BATCH*NVAR + BATCH + rowBase + tid] = lds[OFF_RP + tid] * (1.0f/MAXITER);
  }
}

extern "C" void kernel_launch(void* const* d_in, const int* in_sizes, int n_in,
                              void* d_out, int out_size, void* d_ws, size_t ws_size,
                              hipStream_t stream) {
  (void)in_sizes; (void)n_in; (void)d_ws; (void)ws_size; (void)out_size;
  const float* lamda = (const float*)d_in[0];
  const float* cin   = (const float*)d_in[1];
  const float* csamp = (const float*)d_in[2];
  const float* beq   = (const float*)d_in[3];
  const float* Av    = (const float*)d_in[4];
  const float* Aa    = (const float*)d_in[5];
  const float* Ap    = (const float*)d_in[6];
  const float* Qinv  = (const float*)d_in[7];
  float* out = (float*)d_out;

  const size_t shmem = (size_t)LDS_FLOATS * sizeof(float);  // ~272 KB < 320 KB/WGP
  static bool attrSet = false;   // host-side attribute; not a stream op
  if (!attrSet) {
    (void)hipFuncSetAttribute((const void*)admm_proj_kernel,
                              hipFuncAttributeMaxDynamicSharedMemorySize,
                              (int)shmem);
    attrSet = true;
  }
  admm_proj_kernel<<<BATCH/TM, NTHREADS, shmem, stream>>>(
      lamda, cin, csamp, beq, Av, Aa, Ap, Qinv, out);
}